// Aspect_Category_Prediction_v2_55654186222167
// MI455X (gfx1250) — compile-verified
//
#include <hip/hip_runtime.h>
#include <hip/hip_bf16.h>

#define EMBED   768
#define HID     768
#define KDIM    1536         // EMBED + HID
#define ASPECTS 32
#define BATCH   256
#define KCHUNK  512          // K staging chunk in LDS
#define ASTRIDE (KCHUNK + 2) // pad: stride%64 == 2 -> no LDS bank conflicts
#define NBLOCK  64           // N columns per workgroup (4 waves x 16)

typedef __attribute__((ext_vector_type(2))) float v2f;
typedef __attribute__((ext_vector_type(8))) float v8f;

// ---------------------------------------------------------------------------
// Kernel 1: group samples by aspect.  counts[32], groups[32][256].
// Slot order may vary (atomics) but each sample's arithmetic is independent
// of its slot -> bitwise-deterministic final output.
// ---------------------------------------------------------------------------
__global__ __launch_bounds__(256) void group_kernel(
    const int* __restrict__ aspect_ids,
    int* __restrict__ counts,
    int* __restrict__ groups)
{
    __shared__ int cnt[ASPECTS];
    const int tid = threadIdx.x;
    if (tid < ASPECTS) cnt[tid] = 0;
    __syncthreads();
    if (tid < BATCH) {
        const int a = aspect_ids[tid];
        const int slot = atomicAdd(&cnt[a], 1);
        groups[a * BATCH + slot] = tid;
    }
    __syncthreads();
    if (tid < ASPECTS) counts[tid] = cnt[tid];
}

// ---------------------------------------------------------------------------
// Kernel 2: per-aspect grouped GEMM, layer 1:
//   out1[s, n] = relu( sum_k V[s,k] * W1[aspect, k, n] + b1[aspect, n] )
// Grid: (32 aspects, 12 N-blocks of 64).  Block: 128 threads = 4 waves,
// each wave owns one 16x16 WMMA tile (16 N columns), M-tiles of 16 samples.
// Inner op: v_wmma_f32_16x16x4_f32 (full fp32, D = A*B + C).
// ---------------------------------------------------------------------------
__global__ __launch_bounds__(128) void layer1_kernel(
    const float* __restrict__ vX,
    const float* __restrict__ vH,
    const float* __restrict__ W1,      // [ASPECTS][KDIM][HID]
    const float* __restrict__ b1,      // [ASPECTS][HID]
    const int*  __restrict__ counts,
    const int*  __restrict__ groups,
    float* __restrict__ out1)          // [BATCH][HID]
{
    const int aspect = blockIdx.x;
    const int n0     = blockIdx.y * NBLOCK;
    const int tid    = threadIdx.x;
    const int wave   = tid >> 5;       // 0..3
    const int lane   = tid & 31;

    const int count = counts[aspect];
    if (count == 0) return;            // uniform per block

    __shared__ float Ash[16 * ASTRIDE];

    const float* __restrict__ W1a = W1 + (size_t)aspect * KDIM * HID;

    const int col = lane & 15;         // N within tile / M row of A
    const int hi  = lane >> 4;         // lane-half selects K pair (ISA layout)
    const int n   = n0 + wave * 16 + col;
    const float bias = b1[aspect * HID + n];

    // staging roles: 128 threads fill 16 rows; thread t -> row t>>3, cols (t&7)+8i
    const int srow = tid >> 3;         // 0..15
    const int scol = tid & 7;

    for (int mt = 0; mt * 16 < count; ++mt) {
        v8f acc = {};

        for (int kc = 0; kc < KDIM; kc += KCHUNK) {
            __syncthreads();           // WAR: previous chunk fully consumed
            // ---- stage A chunk: 16 gathered V rows x KCHUNK cols into LDS
            const int m = mt * 16 + srow;
            if (m < count) {
                const int s = groups[aspect * BATCH + m];
                const float* __restrict__ xr = vX + (size_t)s * EMBED;
                const float* __restrict__ hr = vH + (size_t)s * HID;
                for (int c = scol; c < KCHUNK; c += 8) {
                    const int g = kc + c;
                    Ash[srow * ASTRIDE + c] =
                        (g < EMBED) ? xr[g] : hr[g - EMBED];
                }
            } else {
                for (int c = scol; c < KCHUNK; c += 8)
                    Ash[srow * ASTRIDE + c] = 0.0f;
            }
            __syncthreads();

            // ---- K loop over this chunk: one f32 WMMA per 4 K values
            #pragma unroll 4
            for (int k2 = 0; k2 < KCHUNK; k2 += 4) {
                const int ka = k2 + 2 * hi;          // this lane's K pair
                // A frag: lane holds A[M=col][ka], A[col][ka+1] (8B aligned)
                v2f a = *(const v2f*)&Ash[col * ASTRIDE + ka];
                // B frag: lane holds B[K=ka][N=n], B[ka+1][n]
                const size_t kg = (size_t)(kc + ka);
                v2f b;
                b.x = W1a[kg * HID + n];
                b.y = W1a[(kg + 1) * HID + n];
                acc = __builtin_amdgcn_wmma_f32_16x16x4_f32(
                        /*neg_a=*/false, a, /*neg_b=*/false, b,
                        /*c_mod=*/(short)0, acc,
                        /*reuse_a=*/false, /*reuse_b=*/false);
            }
        }

        // ---- epilogue: C/D layout -> VGPR v holds M = v + 8*hi, N = n
        #pragma unroll
        for (int v = 0; v < 8; ++v) {
            const int m = mt * 16 + v + 8 * hi;
            if (m < count) {
                const int s = groups[aspect * BATCH + m];
                const float val = acc[v] + bias;
                out1[(size_t)s * HID + n] = val > 0.0f ? val : 0.0f;
            }
        }
        __syncthreads();               // before restaging next M-tile
    }
}

// ---------------------------------------------------------------------------
// Kernel 3: layer 2 — per-sample (768 x 2) GEMV + bias.  One wave per sample.
// ---------------------------------------------------------------------------
__global__ __launch_bounds__(256) void layer2_kernel(
    const float* __restrict__ out1,
    const int*  __restrict__ aspect_ids,
    const float* __restrict__ W2,      // [ASPECTS][HID][2]
    const float* __restrict__ b2,      // [ASPECTS][2]
    float* __restrict__ logits)        // [BATCH][2]
{
    const int s    = blockIdx.x * 8 + (threadIdx.x >> 5);
    const int lane = threadIdx.x & 31;
    if (s >= BATCH) return;

    const int a = aspect_ids[s];
    const float* __restrict__ o = out1 + (size_t)s * HID;
    const float* __restrict__ w = W2 + (size_t)a * HID * 2;

    float a0 = 0.0f, a1 = 0.0f;
    for (int h = lane; h < HID; h += 32) {
        const float x = o[h];
        a0 += x * w[h * 2 + 0];
        a1 += x * w[h * 2 + 1];
    }
    // wave32 tree reduction
    #pragma unroll
    for (int off = 16; off > 0; off >>= 1) {
        a0 += __shfl_down(a0, off, 32);
        a1 += __shfl_down(a1, off, 32);
    }
    if (lane == 0) {
        logits[s * 2 + 0] = a0 + b2[a * 2 + 0];
        logits[s * 2 + 1] = a1 + b2[a * 2 + 1];
    }
}

// ---------------------------------------------------------------------------
extern "C" void kernel_launch(void* const* d_in, const int* in_sizes, int n_in,
                              void* d_out, int out_size, void* d_ws, size_t ws_size,
                              hipStream_t stream)
{
    const float* vX  = (const float*)d_in[0];   // (256, 768)
    const float* vH  = (const float*)d_in[1];   // (256, 768)
    const int*   ids = (const int*)  d_in[2];   // (256,)
    const float* W1  = (const float*)d_in[3];   // (32, 1536*768)
    const float* b1  = (const float*)d_in[4];   // (32, 768)
    const float* W2  = (const float*)d_in[5];   // (32, 1536)
    const float* b2  = (const float*)d_in[6];   // (32, 2)
    float* logits = (float*)d_out;              // (256, 2)

    char* ws = (char*)d_ws;
    int*   counts = (int*)ws;                                   // 32 ints
    int*   groups = (int*)(ws + 128);                           // 32*256 ints
    float* out1   = (float*)(ws + 128 + ASPECTS * BATCH * 4);   // 256*768 f32

    group_kernel<<<1, 256, 0, stream>>>(ids, counts, groups);

    dim3 g1(ASPECTS, HID / NBLOCK);   // 32 x 12 = 384 workgroups
    layer1_kernel<<<g1, 128, 0, stream>>>(vX, vH, W1, b1, counts, groups, out1);

    layer2_kernel<<<BATCH / 8, 256, 0, stream>>>(out1, ids, W2, b2, logits);
}